// SheafLearning_52982716564306
// MI455X (gfx1250) — compile-verified
//
#include <hip/hip_runtime.h>

typedef float v2f __attribute__((ext_vector_type(2)));
typedef float v4f __attribute__((ext_vector_type(4)));
typedef float v8f __attribute__((ext_vector_type(8)));

#define N_NODES 1024
#define DBLK 8          // sheaf block size D
#define CELL 64         // D*D floats per (src,dst) cell

// ---------------------------------------------------------------------------
// Kernel 1: zero the 256MB dense output with nontemporal b128 stores.
// ---------------------------------------------------------------------------
__global__ void sheaf_zero_out(v4f* __restrict__ out, long long n4) {
  long long i = (long long)blockIdx.x * blockDim.x + threadIdx.x;
  long long stride = (long long)gridDim.x * blockDim.x;
  v4f z = {0.f, 0.f, 0.f, 0.f};
  for (; i < n4; i += stride) __builtin_nontemporal_store(z, &out[i]);
}

// ---------------------------------------------------------------------------
// Kernel 2: init winner array to -1.
// ---------------------------------------------------------------------------
__global__ void sheaf_init_winner(int* __restrict__ w, int n) {
  int i = blockIdx.x * blockDim.x + threadIdx.x;
  int stride = gridDim.x * blockDim.x;
  for (; i < n; i += stride) w[i] = -1;
}

// ---------------------------------------------------------------------------
// Kernel 3: last-write-wins arbitration for duplicate (src,dst) pairs.
// ---------------------------------------------------------------------------
__global__ void sheaf_winner(const int* __restrict__ src,
                             const int* __restrict__ dst,
                             int* __restrict__ winner, int e) {
  int i = blockIdx.x * blockDim.x + threadIdx.x;
  if (i < e) {
    long long cell = (long long)src[i] * N_NODES + dst[i];
    atomicMax(&winner[cell], i);
  }
}

// ---------------------------------------------------------------------------
// Kernel 4: per-wave 16-edge x 64-output tile via V_WMMA_F32_16X16X4_F32.
//   A (16x4 fp32): lanes 0-15 -> row M=lane, K={0,1}; lanes 16-31 -> K={2,3}.
//   B (4x16 fp32): lanes 0-15 -> col N=lane, K={0,1}; lanes 16-31 -> K={2,3}.
//   C/D (16x16 fp32, 8 VGPRs): VGPR v -> M = v + 8*(lane>=16), N = lane&15.
// K=8 handled as two chained K=4 WMMAs. 4 N-tiles cover the 64 outputs.
// Losing duplicate edges store (branchlessly) into a dump region in d_ws.
// ---------------------------------------------------------------------------
__global__ void sheaf_wmma_scatter(const float* __restrict__ x,
                                   const int*   __restrict__ src,
                                   const int*   __restrict__ dst,
                                   const float* __restrict__ Wm,
                                   const int*   __restrict__ winner,
                                   float*       __restrict__ dump,
                                   float*       __restrict__ out,
                                   int e) {
  const int lane = threadIdx.x & 31;
  const int wave = (int)((blockIdx.x * blockDim.x + threadIdx.x) >> 5);
  const int edge_base = wave * 16;
  if (edge_base >= e) return;

  const int half = lane >> 4;   // 0: lanes 0-15, 1: lanes 16-31
  const int mn   = lane & 15;   // A row / B-D column within tile

  // ---- A operand: x_in = |x[src] - x[dst]| for this lane's (row, K-pair)
  int ae = edge_base + mn;
  if (ae >= e) ae = e - 1;                 // clamp (e is a multiple of 16 here)
  const float* xs = x + src[ae] * DBLK;
  const float* xd = x + dst[ae] * DBLK;
  const int k0 = half * 2;
  v2f a0, a1;
  a0.x = fabsf(xs[k0]     - xd[k0]);
  a0.y = fabsf(xs[k0 + 1] - xd[k0 + 1]);
  a1.x = fabsf(xs[k0 + 4] - xd[k0 + 4]);
  a1.y = fabsf(xs[k0 + 5] - xd[k0 + 5]);

  // ---- Per-lane store bases: rows M = v + 8*half; branchless winner steer.
  float* pbase[8];
#pragma unroll
  for (int v = 0; v < 8; ++v) {
    int edg = edge_base + v + 8 * half;
    int ce  = (edg < e) ? edg : e - 1;
    long long cell = (long long)src[ce] * N_NODES + dst[ce];
    bool ok = (edg < e) && (!winner || winner[cell] == edg);
    pbase[v] = ok ? (out + cell * CELL) : dump;   // 64-bit cndmask, no branch
  }

  // ---- 4 N-tiles of 16 columns each
#pragma unroll
  for (int t = 0; t < 4; ++t) {
    const int n_glob = t * 16 + mn;          // output column j in [0,64)
    const float* wr = Wm + n_glob * DBLK;    // W[j, :], row-major [64,8]
    v2f b0, b1;
    b0.x = wr[k0];
    b0.y = wr[k0 + 1];
    b1.x = wr[k0 + 4];
    b1.y = wr[k0 + 5];

    v8f c = {0.f, 0.f, 0.f, 0.f, 0.f, 0.f, 0.f, 0.f};
    // D = A(:,0:4) x B(0:4,:) + A(:,4:8) x B(4:8,:)
    c = __builtin_amdgcn_wmma_f32_16x16x4_f32(false, a0, false, b0,
                                              (short)0, c, false, false);
    c = __builtin_amdgcn_wmma_f32_16x16x4_f32(false, a1, false, b1,
                                              (short)0, c, false, false);

    // Unconditional scatter: each 16-lane half writes 16 contiguous dwords
    // of one cell row (losers land in the dump region).
#pragma unroll
    for (int v = 0; v < 8; ++v) {
      __builtin_nontemporal_store(c[v], pbase[v] + n_glob);
    }
  }
}

// ---------------------------------------------------------------------------
extern "C" void kernel_launch(void* const* d_in, const int* in_sizes, int n_in,
                              void* d_out, int out_size, void* d_ws, size_t ws_size,
                              hipStream_t stream) {
  const float* x   = (const float*)d_in[0];
  const int*   ei  = (const int*)d_in[1];   // edge_index flattened [2, e]
  const float* Wm  = (const float*)d_in[2]; // [64, 8]
  float*       out = (float*)d_out;         // [1024,1024,8,8]

  const int e = in_sizes[1] / 2;
  const int* src = ei;
  const int* dst = ei + e;

  // Workspace layout: [winner: n*n ints = 4MB][dump: 64 floats]
  const size_t winner_bytes = (size_t)N_NODES * N_NODES * sizeof(int);
  const size_t need = winner_bytes + CELL * sizeof(float);
  int*   winner = nullptr;
  float* dump   = nullptr;
  if (ws_size >= need) {
    winner = (int*)d_ws;
    dump   = (float*)((char*)d_ws + winner_bytes);
  } else if (ws_size >= CELL * sizeof(float)) {
    dump = (float*)d_ws;                    // no winner arbitration fallback
  }

  // 1) zero the 256 MB output (bandwidth-bound; ~11 us at 23.3 TB/s)
  {
    long long n4 = (long long)out_size / 4;   // float4 count
    sheaf_zero_out<<<4096, 256, 0, stream>>>((v4f*)out, n4);
  }

  if (winner) {
    // 2) init winner
    sheaf_init_winner<<<2048, 256, 0, stream>>>(winner, N_NODES * N_NODES);
    // 3) last-write-wins arbitration
    sheaf_winner<<<(e + 255) / 256, 256, 0, stream>>>(src, dst, winner, e);
  }

  // 4) WMMA edge GEMM + scatter: one wave per 16 edges
  {
    const int waves  = (e + 15) / 16;
    const int blocks = (waves + 7) / 8;       // 8 waves (256 threads) per block
    sheaf_wmma_scatter<<<blocks, 256, 0, stream>>>(x, src, dst, Wm, winner,
                                                   dump, out, e);
  }
}